// ChamferDistanceLoss_50646254354947
// MI455X (gfx1250) — compile-verified
//
#include <hip/hip_runtime.h>
#include <stdint.h>

typedef __attribute__((ext_vector_type(2))) float v2f;
typedef __attribute__((ext_vector_type(8))) float v8f;

#define BATCH 16
#define NPTS 4096                 // N == M
#define TILES_J (NPTS / 16)       // 256 column tiles
#define ROWS_PER_WAVE 32          // two 16-row WMMA tiles per wave
#define WAVES_PER_DIR (BATCH * (NPTS / ROWS_PER_WAVE))   // 2048
#define TOTAL_WAVES (2 * WAVES_PER_DIR)                  // 4096

// ---------------------------------------------------------------------------
// Pass 1: pack operands for BOTH gemm directions.
// For cloud c:  Apack[c][i] = ( x,  y,  z, 1)       (A rows)
//               Bpack[c][i] = (-2x,-2y,-2z, |p|^2)  (B cols; w carries norm)
//               norm [c][i] = |p|^2                 (C per-row bias)
// dist tile:  D = A_rows(X) * B_cols(Y) + C(normX)  = full 16x16 sq-dist tile
// ---------------------------------------------------------------------------
__global__ void __launch_bounds__(256) chamfer_prep(
    const float* __restrict__ x1, const float* __restrict__ x2,
    float4* __restrict__ A1, float4* __restrict__ B1, float* __restrict__ n1,
    float4* __restrict__ A2, float4* __restrict__ B2, float* __restrict__ n2)
{
    int idx = blockIdx.x * blockDim.x + threadIdx.x;
    if (idx >= BATCH * NPTS) return;

    float ax = x1[idx * 3 + 0], ay = x1[idx * 3 + 1], az = x1[idx * 3 + 2];
    float na = ax * ax + ay * ay + az * az;
    A1[idx] = make_float4(ax, ay, az, 1.0f);
    B1[idx] = make_float4(-2.0f * ax, -2.0f * ay, -2.0f * az, na);
    n1[idx] = na;

    float bx = x2[idx * 3 + 0], by = x2[idx * 3 + 1], bz = x2[idx * 3 + 2];
    float nb = bx * bx + by * by + bz * bz;
    A2[idx] = make_float4(bx, by, bz, 1.0f);
    B2[idx] = make_float4(-2.0f * bx, -2.0f * by, -2.0f * bz, nb);
    n2[idx] = nb;
}

// ---------------------------------------------------------------------------
// Pass 2: one wave32 per (direction, batch, 32-row block).
// Inner loop: 1x global_load_b64 (B tile, prefetched one ahead)
//           + 2x v_wmma_f32_16x16x4_f32 + 16x v_min. No DS, no atomics.
// ---------------------------------------------------------------------------
__global__ void __launch_bounds__(256) chamfer_main(
    const float4* __restrict__ A1, const float4* __restrict__ B1,
    const float*  __restrict__ n1,
    const float4* __restrict__ A2, const float4* __restrict__ B2,
    const float*  __restrict__ n2,
    float* __restrict__ rowMin1, float* __restrict__ rowMin2)
{
    const int lane  = threadIdx.x & 31;
    const int gwave = blockIdx.x * (blockDim.x >> 5) + (threadIdx.x >> 5);
    const int dir   = gwave >> 11;               // 0: x1 rows, 1: x2 rows
    const int widx  = gwave & (WAVES_PER_DIR - 1);
    const int b     = widx >> 7;                 // 128 waves per batch
    const int i0    = (widx & 127) * ROWS_PER_WAVE;

    const float4* Ap   = dir ? A2 : A1;
    const float4* Bp   = dir ? B1 : B2;          // opposite cloud's columns
    const float*  np   = dir ? n2 : n1;
    float*        rOut = dir ? rowMin2 : rowMin1;

    const int half = lane >> 4;                  // 0: K={0,1}  1: K={2,3}
    const int l16  = lane & 15;
    const int k0   = half * 2;

    // A operands (loop invariant): rows i0+l16 and i0+16+l16
    const float* arow0 = (const float*)&Ap[(size_t)b * NPTS + i0 + l16];
    const float* arow1 = arow0 + 16 * 4;
    v2f a0 = *(const v2f*)(arow0 + k0);
    v2f a1 = *(const v2f*)(arow1 + k0);

    // C operands (loop invariant): c[v] = norm of row (i0 + v + 8*half)
    const float* npp = np + (size_t)b * NPTS + i0 + 8 * half;
    v8f c0, c1;
#pragma unroll
    for (int v = 0; v < 8; ++v) { c0[v] = npp[v]; c1[v] = npp[16 + v]; }

    v8f r0, r1;
#pragma unroll
    for (int v = 0; v < 8; ++v) { r0[v] = __builtin_inff(); r1[v] = __builtin_inff(); }

    const float* bbase = (const float*)&Bp[(size_t)b * NPTS];

    // rotating prefetch: bb holds tile jt, bn fetches tile jt+1 (wraps, valid)
    v2f bb = *(const v2f*)(bbase + (size_t)(l16 * 4) + k0);

#pragma unroll 4
    for (int jt = 0; jt < TILES_J; ++jt) {
        int jn = (jt + 1) & (TILES_J - 1);
        v2f bn = *(const v2f*)(bbase + (size_t)((jn * 16 + l16) * 4) + k0);

        v8f d0 = __builtin_amdgcn_wmma_f32_16x16x4_f32(
            false, a0, false, bb, (short)0, c0, false, false);
        v8f d1 = __builtin_amdgcn_wmma_f32_16x16x4_f32(
            false, a1, false, bb, (short)0, c1, false, false);

#pragma unroll
        for (int v = 0; v < 8; ++v) {
            r0[v] = fminf(r0[v], d0[v]);
            r1[v] = fminf(r1[v], d1[v]);
        }
        bb = bn;
    }

    // Row-min epilogue: reduce across the 16 n-lanes of each half.
#pragma unroll
    for (int v = 0; v < 8; ++v) {
        float x = r0[v], y = r1[v];
        x = fminf(x, __shfl_xor(x, 1, 32));
        y = fminf(y, __shfl_xor(y, 1, 32));
        x = fminf(x, __shfl_xor(x, 2, 32));
        y = fminf(y, __shfl_xor(y, 2, 32));
        x = fminf(x, __shfl_xor(x, 4, 32));
        y = fminf(y, __shfl_xor(y, 4, 32));
        x = fminf(x, __shfl_xor(x, 8, 32));
        y = fminf(y, __shfl_xor(y, 8, 32));
        if (l16 == 0) {
            size_t base = (size_t)b * NPTS + i0 + v + 8 * half;
            rOut[base]      = x;
            rOut[base + 16] = y;
        }
    }
}

// ---------------------------------------------------------------------------
// Pass 3: per-batch mean of both row-min arrays.
// ---------------------------------------------------------------------------
__global__ void __launch_bounds__(256) chamfer_final(
    const float* __restrict__ rowMin1, const float* __restrict__ rowMin2,
    float* __restrict__ out)
{
    const int b = blockIdx.x;
    __shared__ float s1[256];
    __shared__ float s2[256];

    float a = 0.0f, c = 0.0f;
    for (int i = threadIdx.x; i < NPTS; i += 256) {
        a += rowMin1[(size_t)b * NPTS + i];
        c += rowMin2[(size_t)b * NPTS + i];
    }
    s1[threadIdx.x] = a;
    s2[threadIdx.x] = c;
    __syncthreads();
    for (int s = 128; s > 0; s >>= 1) {
        if (threadIdx.x < s) {
            s1[threadIdx.x] += s1[threadIdx.x + s];
            s2[threadIdx.x] += s2[threadIdx.x + s];
        }
        __syncthreads();
    }
    if (threadIdx.x == 0)
        out[b] = s1[0] / (float)NPTS + s2[0] / (float)NPTS;
}

// ---------------------------------------------------------------------------
extern "C" void kernel_launch(void* const* d_in, const int* in_sizes, int n_in,
                              void* d_out, int out_size, void* d_ws, size_t ws_size,
                              hipStream_t stream)
{
    const float* x1 = (const float*)d_in[0];   // [B, N, 3]
    const float* x2 = (const float*)d_in[1];   // [B, M, 3]
    float* out = (float*)d_out;                // [B]

    // Workspace layout (~5 MB)
    char* ws = (char*)d_ws;
    float4* A1 = (float4*)(ws + 0);                       // 1 MiB
    float4* B1 = (float4*)(ws + (1u << 20));              // 1 MiB
    float4* A2 = (float4*)(ws + (2u << 20));              // 1 MiB
    float4* B2 = (float4*)(ws + (3u << 20));              // 1 MiB
    float*  n1 = (float*) (ws + (4u << 20));              // 256 KiB
    float*  n2 = (float*) (ws + (4u << 20) + 262144);     // 256 KiB
    float*  r1 = (float*) (ws + (4u << 20) + 2 * 262144); // 256 KiB
    float*  r2 = (float*) (ws + (4u << 20) + 3 * 262144); // 256 KiB

    chamfer_prep<<<(BATCH * NPTS + 255) / 256, 256, 0, stream>>>(
        x1, x2, A1, B1, n1, A2, B2, n2);

    // 4096 waves (both directions), 8 waves per block
    chamfer_main<<<TOTAL_WAVES / 8, 256, 0, stream>>>(
        A1, B1, n1, A2, B2, n2, r1, r2);

    chamfer_final<<<BATCH, 256, 0, stream>>>(r1, r2, out);
}